// Simulator_65678639890850
// MI455X (gfx1250) — compile-verified
//
#include <hip/hip_runtime.h>
#include <math.h>

// ---------------- constants from the reference ----------------
#define N_PED      4096
#define DT_STEP    0.04f            // 0.4 / 10
#define DTF        0.4f             // DELTA_T_FULL
#define V0C        2.1f
#define INV_SIGMA  (1.0f/0.3f)
#define COSPHI_C   (-0.17364817766693041f)   // cos(100 deg)
#define OOV        0.5f
#define FD         0.001f
#define INV_FD     1000.0f
#define MSM        1.3f
#define NSTEPS     10

// ---------------- tiling ----------------
#define TB          256                  // threads per block (8 waves)
#define A_PER_BLOCK 128                  // 8 waves x 16 a-rows
#define CHUNK_B     256                  // b-peds per block (8KB LDS aux tile)
#define NCHUNK      (N_PED/CHUNK_B)      // 16
#define ABLOCKS     (N_PED/A_PER_BLOCK)  // 32

typedef float v2f __attribute__((ext_vector_type(2)));
typedef float v8f __attribute__((ext_vector_type(8)));

#if __has_builtin(__builtin_amdgcn_wmma_f32_16x16x4_f32)
#define HAVE_WMMA_F32_K4 1
#else
#define HAVE_WMMA_F32_K4 0
#endif

// ---------- CDNA5 async global->LDS staging (gfx1250) ----------
__device__ __forceinline__ void async_load_b128(unsigned lds_addr, const void* gaddr) {
  asm volatile("global_load_async_to_lds_b128 %0, %1, off"
               :: "v"(lds_addr), "v"(gaddr)
               : "memory");
}
__device__ __forceinline__ void wait_asynccnt0() {
  asm volatile("s_wait_asynccnt 0" ::: "memory");
}

__device__ __forceinline__ float pair_value(float nsum, float c) {
  float insq = fmaxf(nsum * nsum - c, 0.0f);
  float bb   = 0.5f * __builtin_amdgcn_sqrtf(insq);
  return V0C * __expf(-bb * INV_SIGMA);
}

// ---------------- kernels ----------------

// Copy input state into working state (d_out) and fix desired/max speeds
// (computed ONCE from the initial velocities, per the reference).
__global__ void sf_init(const float* __restrict__ in, float* __restrict__ state,
                        float* __restrict__ dspd, float* __restrict__ mspd) {
  int i = blockIdx.x * blockDim.x + threadIdx.x;
  if (i < N_PED * 9) state[i] = in[i];
  if (i < N_PED) {
    float vx = in[i * 9 + 2], vy = in[i * 9 + 3];
    float sp = __builtin_amdgcn_sqrtf(vx * vx + vy * vy);
    dspd[i] = sp;
    mspd[i] = MSM * sp;
  }
}

// Per-step per-ped aux: [p.x, p.y, q.x, q.y, (0.4*speed)^2, pad x3]
// (q = p + 0.4*speed*e).  Also zeroes the force accumulator.
__global__ void sf_aux(const float* __restrict__ state, float* __restrict__ aux,
                       float* __restrict__ Facc) {
  int i = blockIdx.x * blockDim.x + threadIdx.x;
  if (i >= N_PED) return;
  const float* s = state + i * 9;
  float px = s[0], py = s[1];
  float vx = s[2], vy = s[3];
  float spd = __builtin_amdgcn_sqrtf(vx * vx + vy * vy);
  float dx = s[6] - px, dy = s[7] - py;
  float inv = 1.0f / __builtin_amdgcn_sqrtf(dx * dx + dy * dy);
  float ds = DTF * spd;
  float* o = aux + i * 8;
  o[0] = px; o[1] = py;
  o[2] = px + ds * dx * inv; o[3] = py + ds * dy * inv;
  o[4] = ds * ds; o[5] = 0.0f; o[6] = 0.0f; o[7] = 0.0f;
  Facc[2 * i] = 0.0f;
  Facc[2 * i + 1] = 0.0f;
}

// Pairwise social force.
//   Block: 128 'a' rows (16 per wave) x 256 'b' columns (LDS aux tile, async-staged).
//   Per 16x16 tile, one V_WMMA_F32_16X16X4_F32 computes the FOV dot matrix
//   Dep[m][n] = e_a(m) . p_b(n) (K=2 padded to 4); its C/D layout (lane = col n,
//   VGPR g = row 8*(lane/16)+g) defines which 8 pairs each lane then processes.
//   Distances are computed directly from rx,ry (no Gram-matrix cancellation).
__global__ void __launch_bounds__(TB)
sf_force(const float* __restrict__ state, const float* __restrict__ aux,
         float* __restrict__ Facc) {
  __shared__ float smem[CHUNK_B * 8];          // b-side aux tile (8KB)
  __shared__ float rowtab[A_PER_BLOCK * 4];    // a-side rows: {px, py, ex, ey}

  const int tid   = threadIdx.x;
  const int wv    = tid >> 5;                  // wave id 0..7
  const int lane  = tid & 31;
  const int hf    = lane >> 4;                 // 0: rows m=0..7 | 1: rows m=8..15
  const int n     = lane & 15;                 // column within 16-wide b tile
  const int aBase = blockIdx.x * A_PER_BLOCK;
  const int b0    = blockIdx.y * CHUNK_B;

  // Stage CHUNK_B aux records (8KB) via CDNA5 async B128 transfers: 2/thread.
  {
    const float* gsrc = aux + (size_t)b0 * 8;
    for (int k = tid; k < CHUNK_B * 2; k += TB) {
      unsigned lds = (unsigned)(unsigned long long)(const void*)&smem[k * 4];
      async_load_b128(lds, gsrc + k * 4);
    }
  }
  // Build the a-row table while the async DMA is in flight.
  if (tid < A_PER_BLOCK) {
    const float* s = state + (size_t)(aBase + tid) * 9;
    float px = s[0], py = s[1];
    float dx = s[6] - px, dy = s[7] - py;
    float inv = 1.0f / __builtin_amdgcn_sqrtf(dx * dx + dy * dy);
    float* rr = &rowtab[tid * 4];
    rr[0] = px; rr[1] = py; rr[2] = dx * inv; rr[3] = dy * inv;
  }
  wait_asynccnt0();
  __syncthreads();

  // Preload this lane's 8 a-rows into registers (no LDS in the hot loop).
  float r_pax[8], r_pay[8], r_epa[8];
#if !HAVE_WMMA_F32_K4
  float r_eax[8], r_eay[8];
#endif
  const int mloc = wv * 16 + 8 * hf;
#pragma unroll
  for (int g = 0; g < 8; ++g) {
    const float* rr = &rowtab[(mloc + g) * 4];
    float px = rr[0], py = rr[1], ex = rr[2], ey = rr[3];
    r_pax[g] = px; r_pay[g] = py; r_epa[g] = ex * px + ey * py;
#if !HAVE_WMMA_F32_K4
    r_eax[g] = ex; r_eay[g] = ey;
#endif
  }

  // A operand (16x4 f32): lanes 0-15 rows M=0..15, VGPR0=K0, VGPR1=K1;
  // lanes 16-31 carry K=2,3 which we zero.
  v2f Ae = {0.0f, 0.0f};
  if (hf == 0) {
    const float* rr = &rowtab[(wv * 16 + n) * 4];
    Ae.x = rr[2]; Ae.y = rr[3];
  }

  float accx[8], accy[8];
#pragma unroll
  for (int g = 0; g < 8; ++g) { accx[g] = 0.0f; accy[g] = 0.0f; }

  for (int bt = 0; bt < CHUNK_B / 16; ++bt) {
    // Column (b-side) scalars for this lane's column n.
    const float* col = &smem[(bt * 16 + n) * 8];
    float pbx = col[0], pby = col[1], qbx = col[2], qby = col[3], cb = col[4];

#if HAVE_WMMA_F32_K4
    // B operand (4x16 f32): lanes 0-15 cols N=0..15, VGPR0=K0, VGPR1=K1;
    // lanes 16-31 carry K=2,3 -> zero.
    v2f Bp = {0.0f, 0.0f};
    if (hf == 0) { Bp.x = pbx; Bp.y = pby; }
    v8f cz = {0.f, 0.f, 0.f, 0.f, 0.f, 0.f, 0.f, 0.f};
    v8f dep = __builtin_amdgcn_wmma_f32_16x16x4_f32(
        false, Ae, false, Bp, (short)0, cz, false, false);
#else
    (void)Ae;
#endif

    const int bcol = b0 + bt * 16 + n;
#pragma unroll
    for (int g = 0; g < 8; ++g) {
      float pax = r_pax[g], pay = r_pay[g];
      float rx = pax - pbx, ry = pay - pby;     // r_ab (exact)
      float sx = pax - qbx, sy = pay - qby;     // shifted (exact)

      float nr0 = __builtin_amdgcn_sqrtf(rx * rx + ry * ry);
      float ns0 = __builtin_amdgcn_sqrtf(sx * sx + sy * sy);
      float v0  = pair_value(nr0 + ns0, cb);

      float rx1 = rx + FD, sx1 = sx + FD;
      float nr1 = __builtin_amdgcn_sqrtf(rx1 * rx1 + ry * ry);
      float ns1 = __builtin_amdgcn_sqrtf(sx1 * sx1 + sy * sy);
      float v1  = pair_value(nr1 + ns1, cb);

      float ry2 = ry + FD, sy2 = sy + FD;
      float nr2 = __builtin_amdgcn_sqrtf(rx * rx + ry2 * ry2);
      float ns2 = __builtin_amdgcn_sqrtf(sx * sx + sy2 * sy2);
      float v2  = pair_value(nr2 + ns2, cb);

      // FOV: e_a . r_ab = e_a.p_a - e_a.p_b (Dep from the WMMA tile)
#if HAVE_WMMA_F32_K4
      float dot = r_epa[g] - dep[g];
#else
      float dot = r_eax[g] * rx + r_eay[g] * ry;
#endif
      float w = (dot > nr0 * COSPHI_C) ? 1.0f : OOV;
      if (bcol == aBase + mloc + g) w = 0.0f;   // diagonal

      accx[g] += w * (v0 - v1);
      accy[g] += w * (v0 - v2);
    }
  }

  // Reduce each row's partial over the 16 column-lanes of its half, then merge.
#pragma unroll
  for (int g = 0; g < 8; ++g) {
    float fx = accx[g] * INV_FD, fy = accy[g] * INV_FD;
    for (int m = 1; m < 16; m <<= 1) {
      fx += __shfl_xor(fx, m, 32);
      fy += __shfl_xor(fy, m, 32);
    }
    if (n == 0) {
      int a = aBase + mloc + g;
      atomicAdd(&Facc[2 * a],     fx);
      atomicAdd(&Facc[2 * a + 1], fy);
    }
  }
}

// Integrate one step in place (each thread owns its row; cross-ped data
// was consumed from the aux snapshot, so in-place update is race-free).
__global__ void sf_integrate(float* __restrict__ state, const float* __restrict__ Facc,
                             const float* __restrict__ dspd, const float* __restrict__ mspd) {
  int i = blockIdx.x * blockDim.x + threadIdx.x;
  if (i >= N_PED) return;
  float* s = state + i * 9;
  float px = s[0], py = s[1], vx = s[2], vy = s[3];
  float ax = s[4], ay = s[5], dex = s[6], dey = s[7], tau = s[8];

  float ddx = dex - px, ddy = dey - py;
  float inv = 1.0f / __builtin_amdgcn_sqrtf(ddx * ddx + ddy * ddy);
  float ex = ddx * inv, ey = ddy * inv;

  float invtau = 1.0f / tau;
  float Fx = (dspd[i] * ex - vx) * invtau + Facc[2 * i];
  float Fy = (dspd[i] * ey - vy) * invtau + Facc[2 * i + 1];

  float npx = px + vx * DT_STEP + 0.5f * Fx * DT_STEP * DT_STEP;
  float npy = py + vy * DT_STEP + 0.5f * Fy * DT_STEP * DT_STEP;
  float nvx = vx + 0.5f * (ax + Fx) * DT_STEP;
  float nvy = vy + 0.5f * (ay + Fy) * DT_STEP;

  float spd = __builtin_amdgcn_sqrtf(nvx * nvx + nvy * nvy);
  float f = fminf(mspd[i] / spd, 1.0f);
  s[0] = npx; s[1] = npy;
  s[2] = nvx * f; s[3] = nvy * f;
  s[4] = Fx; s[5] = Fy;
  // s[6..8] (dest, tau) unchanged
}

// ---------------- host launcher ----------------
extern "C" void kernel_launch(void* const* d_in, const int* in_sizes, int n_in,
                              void* d_out, int out_size, void* d_ws, size_t ws_size,
                              hipStream_t stream) {
  (void)in_sizes; (void)n_in; (void)out_size; (void)ws_size;
  const float* in = (const float*)d_in[0];

  float* state = (float*)d_out;             // working state = output buffer (N*9)
  float* aux   = (float*)d_ws;              // 8N floats
  float* Facc  = aux + 8 * N_PED;           // 2N floats
  float* dspd  = Facc + 2 * N_PED;          // N floats
  float* mspd  = dspd + N_PED;              // N floats   (total 12N*4 = 192 KB)

  sf_init<<<(N_PED * 9 + 255) / 256, 256, 0, stream>>>(in, state, dspd, mspd);

  for (int t = 0; t < NSTEPS; ++t) {
    sf_aux<<<(N_PED + 255) / 256, 256, 0, stream>>>(state, aux, Facc);
    dim3 grid(ABLOCKS, NCHUNK);
    sf_force<<<grid, TB, 0, stream>>>(state, aux, Facc);
    sf_integrate<<<(N_PED + 255) / 256, 256, 0, stream>>>(state, Facc, dspd, mspd);
  }
}